// HartleySpectralConvV4_55473797595664
// MI455X (gfx1250) — compile-verified
//
#include <hip/hip_runtime.h>

// ---------------------------------------------------------------------------
// Hartley spectral conv on MI455X (gfx1250), all GEMMs via V_WMMA_F32_16X16X4_F32.
//
// Sizes: B=16, CI=CO=64, H=W=256, modes 32 -> packed 64x64 corner spectrum.
// Forward separable DHT computed at 80 padded modes {0..39, 216..255} which
// covers the needed {0..32, 224..255} (incl. reversal partner of k=224 -> 32).
// True 2D DHT recovered with: H(k1,k2) = 0.5*[T(+,+)+T(-,+)+T(+,-)-T(-,-)].
// Inverse: spectrum pre-combined with the same identity, then separable
// inverse with the same 80-mode cas basis, scaled by 0.5/65536.
//
// d_ws usage: ~218 MB (fits comfortably; intermediates are L2-resident).
// ---------------------------------------------------------------------------

typedef __attribute__((ext_vector_type(2))) float v2f;
typedef __attribute__((ext_vector_type(8))) float v8f;

__device__ __forceinline__ v8f wmma4(v2f a, v2f b, v8f c) {
  // D(16x16) = A(16x4) * B(4x16) + C   (fp32, wave32)
  return __builtin_amdgcn_wmma_f32_16x16x4_f32(
      /*neg_a=*/false, a, /*neg_b=*/false, b,
      /*c_mod=*/(short)0, c, /*reuse_a=*/false, /*reuse_b=*/false);
}

// A-frag: row-major A, tile at (m0,k0). ISA layout: lane l holds row m0+(l&15),
// VGPR v holds K = k0 + 2*(l>>4) + v  -> contiguous float pair.
__device__ __forceinline__ v2f afrag(const float* A, int lda, int m0, int k0, int lane) {
  int m = m0 + (lane & 15);
  int k = k0 + ((lane >> 4) << 1);
  v2f r;
  r.x = A[m * lda + k];
  r.y = A[m * lda + k + 1];
  return r;
}

// B-frag: row-major B, tile at (k0,n0). Row-striped like C/D: lane l holds
// col n0+(l&15), VGPR v holds row K = k0 + 2*(l>>4) + v.
__device__ __forceinline__ v2f bfrag(const float* B, int ldb, int k0, int n0, int lane) {
  int n = n0 + (lane & 15);
  int k = k0 + ((lane >> 4) << 1);
  v2f r;
  r.x = B[k * ldb + n];
  r.y = B[(k + 1) * ldb + n];
  return r;
}

// C/D layout: VGPR v holds row m0 + v + 8*(l>>4), col n0+(l&15).
__device__ __forceinline__ void store_tile(float* C, int ldc, int m0, int n0, int lane, v8f acc) {
  int n = n0 + (lane & 15);
  int mb = m0 + ((lane >> 4) << 3);
#pragma unroll
  for (int v = 0; v < 8; ++v) C[(mb + v) * ldc + n] = acc[v];
}

__device__ __forceinline__ float casv(int prod) {
  float ang = (float)(prod & 255) * 0.024543692606170259f;  // 2*pi/256
  float s, c;
  sincosf(ang, &s, &c);
  return c + s;
}

// ---------------- basis: Cw[256][80], Cw[n][j] = cas(2*pi*n*kj/256) ----------
__global__ void k_basis(float* __restrict__ Cw) {
  int n = blockIdx.x;  // 0..255
  for (int j = threadIdx.x; j < 80; j += blockDim.x) {
    int k = (j < 40) ? j : j + 176;
    Cw[n * 80 + j] = casv(n * k);
  }
}

// ------------- weight transpose: WT[p][i*64+o] = w[(i*64+o)*4096 + p] -------
__global__ void k_transpose_w(const float* __restrict__ w, float* __restrict__ wt) {
  __shared__ float tile[32][33];
  int pb = blockIdx.x * 32, iob = blockIdx.y * 32;
  for (int r = threadIdx.y; r < 32; r += 8)
    tile[r][threadIdx.x] = w[(long)(iob + r) * 4096 + pb + threadIdx.x];
  __syncthreads();
  for (int r = threadIdx.y; r < 32; r += 8)
    wt[(long)(pb + r) * 4096 + iob + threadIdx.x] = tile[threadIdx.x][r];
}

// ------------- stage A: T1(M x 80) = X(M x 256) * Cw(256 x 80) --------------
// M = 16*64*256 rows (one row per image line). Cw staged in LDS (80 KB).
__global__ __launch_bounds__(256) void k_fwd_w(const float* __restrict__ X,
                                               const float* __restrict__ Cw,
                                               float* __restrict__ T1) {
  extern __shared__ float lds[];  // 256*80 floats
  for (int t = threadIdx.x; t < 5120; t += 256)
    ((float4*)lds)[t] = ((const float4*)Cw)[t];
  __syncthreads();
  int lane = threadIdx.x & 31, wave = threadIdx.x >> 5;
  long m0 = ((long)blockIdx.x * 8 + wave) * 16;
  const float* Arow = X + m0 * 256;
  v8f acc[5];
#pragma unroll
  for (int t = 0; t < 5; ++t) acc[t] = {};
  for (int k0 = 0; k0 < 256; k0 += 4) {
    __builtin_prefetch(Arow + (lane & 15) * 256 + k0 + 32, 0, 1);
    v2f a = afrag(Arow, 256, 0, k0, lane);
#pragma unroll
    for (int t = 0; t < 5; ++t) {
      v2f b = bfrag(lds, 80, k0, t * 16, lane);
      acc[t] = wmma4(a, b, acc[t]);
    }
  }
  float* Crow = T1 + m0 * 80;
#pragma unroll
  for (int t = 0; t < 5; ++t) store_tile(Crow, 80, 0, t * 16, lane, acc[t]);
}

// ------ stage B: per image Sx(80x80) = Cw^T(80x256) * T1_img(256x80) --------
__global__ __launch_bounds__(256) void k_fwd_h(const float* __restrict__ T1,
                                               const float* __restrict__ Cw,
                                               float* __restrict__ Sx) {
  extern __shared__ float lds[];
  float* sCw = lds;            // 256*80
  float* sT1 = lds + 20480;    // 256*80
  long img = blockIdx.x;
  const float* T1i = T1 + img * 20480;
  for (int t = threadIdx.x; t < 5120; t += 256) {
    ((float4*)sCw)[t] = ((const float4*)Cw)[t];
    ((float4*)sT1)[t] = ((const float4*)T1i)[t];
  }
  __syncthreads();
  int lane = threadIdx.x & 31, wave = threadIdx.x >> 5;
  float* Sxi = Sx + img * 6400;
  for (int tile = wave; tile < 25; tile += 8) {
    int i0 = (tile / 5) * 16, j0 = (tile % 5) * 16;
    v8f acc = {};
    for (int k0 = 0; k0 < 256; k0 += 4) {
      v2f a;  // A = Cw^T : A[i][h] = Cw[h*80 + i]
      {
        int m = i0 + (lane & 15);
        int k = k0 + ((lane >> 4) << 1);
        a.x = sCw[k * 80 + m];
        a.y = sCw[(k + 1) * 80 + m];
      }
      v2f b = bfrag(sT1, 80, k0, j0, lane);
      acc = wmma4(a, b, acc);
    }
    store_tile(Sxi, 80, i0, j0, lane, acc);
  }
}

// mode-index helpers: packed u (0..63) -> 80-mode index of F(u) and of -F(u)
__device__ __forceinline__ int jdir(int u) { return u < 32 ? u : u + 16; }
__device__ __forceinline__ int jrev(int u) {
  if (u == 0) return 0;
  if (u < 32) return 80 - u;
  if (u == 32) return 32;
  return 64 - u;
}

// ---- build even/odd activations per spectral position p = u*64+v -----------
// x_even = 0.5*(T(-,+)+T(+,-)),  x_odd = 0.5*(T(+,+)-T(-,-))
__global__ void k_gather_eo(const float* __restrict__ Sx, float* __restrict__ Ae,
                            float* __restrict__ Ao) {
  int p = blockIdx.x;
  int u = p >> 6, v = p & 63;
  int du = jdir(u), dv = jdir(v), ru = jrev(u), rv = jrev(v);
  for (int bi = threadIdx.x; bi < 1024; bi += blockDim.x) {
    const float* S = Sx + (long)bi * 6400;
    float tdd = S[du * 80 + dv];
    float trd = S[ru * 80 + dv];
    float tdr = S[du * 80 + rv];
    float trr = S[ru * 80 + rv];
    Ae[(long)p * 1024 + bi] = 0.5f * (trd + tdr);
    Ao[(long)p * 1024 + bi] = 0.5f * (tdd - trr);
  }
}

// ---- mixing: per p: Out(16x64) = Ae_p(16x64)*WT[p] + Ao_p(16x64)*WT[pbar] --
// Block handles 2 positions; 4 waves per position (one n-tile each).
__global__ __launch_bounds__(256) void k_mix(const float* __restrict__ Ae,
                                             const float* __restrict__ Ao,
                                             const float* __restrict__ WT,
                                             float* __restrict__ Sy) {
  extern __shared__ float lds[];  // 2 * 10240 floats
  int half = threadIdx.x >> 7;
  int p = blockIdx.x * 2 + half;
  int u = p >> 6, v = p & 63;
  int pbar = (((64 - u) & 63) << 6) | ((64 - v) & 63);
  float* s = lds + half * 10240;
  float* sAe = s;
  float* sAo = s + 1024;
  float* sWd = s + 2048;
  float* sWr = s + 6144;
  int tl = threadIdx.x & 127;
  for (int t = tl; t < 256; t += 128) {
    ((float4*)sAe)[t] = ((const float4*)(Ae + (long)p * 1024))[t];
    ((float4*)sAo)[t] = ((const float4*)(Ao + (long)p * 1024))[t];
  }
  for (int t = tl; t < 1024; t += 128) {
    ((float4*)sWd)[t] = ((const float4*)(WT + (long)p * 4096))[t];
    ((float4*)sWr)[t] = ((const float4*)(WT + (long)pbar * 4096))[t];
  }
  __syncthreads();
  int lane = threadIdx.x & 31;
  int nq = (threadIdx.x >> 5) & 3;  // n-tile 0..3
  v8f acc = {};
  for (int k0 = 0; k0 < 64; k0 += 4) {
    v2f a = afrag(sAe, 64, 0, k0, lane);
    v2f b = bfrag(sWd, 64, k0, nq * 16, lane);
    acc = wmma4(a, b, acc);
  }
  for (int k0 = 0; k0 < 64; k0 += 4) {
    v2f a = afrag(sAo, 64, 0, k0, lane);
    v2f b = bfrag(sWr, 64, k0, nq * 16, lane);
    acc = wmma4(a, b, acc);
  }
  store_tile(Sy + (long)p * 1024, 64, 0, nq * 16, lane, acc);  // Sy[p][b*64+o]
}

// ---- spectrum combination for separable inverse ----------------------------
// Sp[img][j1][j2] = (0.5/65536) * [S(f1,f2)+S(-f1,f2)+S(f1,-f2)-S(-f1,-f2)]
__device__ __forceinline__ float fetchS(const float* Sy, int img, int f1, int f2) {
  int p1 = (f1 < 32) ? f1 : ((f1 >= 224) ? f1 - 192 : -1);
  int p2 = (f2 < 32) ? f2 : ((f2 >= 224) ? f2 - 192 : -1);
  if ((p1 | p2) < 0) return 0.0f;
  return Sy[(long)(p1 * 64 + p2) * 1024 + img];
}

__global__ void k_spec_pad(const float* __restrict__ Sy, float* __restrict__ Sp) {
  int img = blockIdx.x;  // b*64+o
  float* o = Sp + (long)img * 6400;
  for (int t = threadIdx.x; t < 6400; t += blockDim.x) {
    int j1 = t / 80, j2 = t % 80;
    int f1 = (j1 < 40) ? j1 : j1 + 176;
    int f2 = (j2 < 40) ? j2 : j2 + 176;
    int n1 = (256 - f1) & 255, n2 = (256 - f2) & 255;
    float s = fetchS(Sy, img, f1, f2) + fetchS(Sy, img, n1, f2) +
              fetchS(Sy, img, f1, n2) - fetchS(Sy, img, n1, n2);
    o[t] = s * (0.5f / 65536.0f);
  }
}

// ---- inverse stage 1: T3(256x80) = Cw(256x80) * Sp_img(80x80) --------------
__global__ __launch_bounds__(256) void k_inv_h(const float* __restrict__ Sp,
                                               const float* __restrict__ Cw,
                                               float* __restrict__ T3) {
  extern __shared__ float lds[];
  float* sCw = lds;          // 256*80
  float* sS = lds + 20480;   // 80*80
  long img = blockIdx.x;
  const float* Si = Sp + img * 6400;
  for (int t = threadIdx.x; t < 5120; t += 256)
    ((float4*)sCw)[t] = ((const float4*)Cw)[t];
  for (int t = threadIdx.x; t < 1600; t += 256)
    ((float4*)sS)[t] = ((const float4*)Si)[t];
  __syncthreads();
  int lane = threadIdx.x & 31, wave = threadIdx.x >> 5;
  float* T3i = T3 + img * 20480;
  for (int tile = wave; tile < 80; tile += 8) {
    int m0 = (tile / 5) * 16, n0 = (tile % 5) * 16;
    v8f acc = {};
    for (int k0 = 0; k0 < 80; k0 += 4) {
      v2f a = afrag(sCw, 80, m0, k0, lane);
      v2f b = bfrag(sS, 80, k0, n0, lane);
      acc = wmma4(a, b, acc);
    }
    store_tile(T3i, 80, m0, n0, lane, acc);
  }
}

// ---- inverse stage 2: Out(256x256) = T3(256x80) * Cw^T(80x256) + bias ------
__global__ __launch_bounds__(256) void k_inv_w(const float* __restrict__ T3,
                                               const float* __restrict__ Cw,
                                               const float* __restrict__ bias,
                                               float* __restrict__ Out) {
  extern __shared__ float lds[];
  float* sCwT = lds;          // 80*256 : CwT[k][n] = Cw[n*80+k]
  float* sT3 = lds + 20480;   // 256*80
  long img = blockIdx.x >> 1;
  int nh = blockIdx.x & 1;
  const float* T3i = T3 + img * 20480;
  for (int t = threadIdx.x; t < 5120; t += 256)
    ((float4*)sT3)[t] = ((const float4*)T3i)[t];
  for (int t = threadIdx.x; t < 20480; t += 256) {
    int n = t / 80, k = t % 80;
    sCwT[k * 256 + n] = Cw[t];
  }
  __syncthreads();
  float bo = bias[(int)(img & 63)];
  int lane = threadIdx.x & 31, wave = threadIdx.x >> 5;
  float* Oi = Out + img * 65536;
  for (int tile = wave; tile < 128; tile += 8) {
    int m0 = (tile >> 3) * 16, n0 = nh * 128 + (tile & 7) * 16;
    v8f acc = {};
    for (int k0 = 0; k0 < 80; k0 += 4) {
      v2f a = afrag(sT3, 80, m0, k0, lane);
      v2f b = bfrag(sCwT, 256, k0, n0, lane);
      acc = wmma4(a, b, acc);
    }
#pragma unroll
    for (int vv = 0; vv < 8; ++vv) acc[vv] += bo;
    store_tile(Oi, 256, m0, n0, lane, acc);
  }
}

// ---------------------------------------------------------------------------
extern "C" void kernel_launch(void* const* d_in, const int* in_sizes, int n_in,
                              void* d_out, int out_size, void* d_ws, size_t ws_size,
                              hipStream_t stream) {
  (void)in_sizes; (void)n_in; (void)out_size; (void)ws_size;
  const float* x = (const float*)d_in[0];      // (16,64,256,256)
  const float* w = (const float*)d_in[1];      // (64,64,64,64)
  const float* bias = (const float*)d_in[2];   // (64,)
  float* out = (float*)d_out;                  // (16,64,256,256)

  float* ws = (float*)d_ws;
  float* Cw = ws;                    //    20480 f : cas basis 256x80
  float* WT = Cw + 20480;            // 16777216 f : weight^T [p][i*64+o]
  float* Ae = WT + 16777216;         //  4194304 f : even act [p][b*64+i]
  float* Ao = Ae + 4194304;          //  4194304 f : odd  act [p][b*64+i]
  float* Sx = Ao + 4194304;          //  6553600 f : separable spectra / Sp
  float* Sy = Sx + 6553600;          //  4194304 f : mixed spectrum [p][b*64+o]
  float* T  = Sy + 4194304;          // 20971520 f : T1 / T3 staging
  // total 56,905,728 floats = ~218 MB of d_ws

  k_basis<<<256, 128, 0, stream>>>(Cw);
  k_transpose_w<<<dim3(128, 128), dim3(32, 8), 0, stream>>>(w, WT);
  k_fwd_w<<<2048, 256, 81920, stream>>>(x, Cw, T);
  k_fwd_h<<<1024, 256, 163840, stream>>>(T, Cw, Sx);
  k_gather_eo<<<4096, 256, 0, stream>>>(Sx, Ae, Ao);
  k_mix<<<2048, 256, 81920, stream>>>(Ae, Ao, WT, Sy);
  k_spec_pad<<<1024, 256, 0, stream>>>(Sy, Sx);           // Sx reused as Sp
  k_inv_h<<<1024, 256, 107520, stream>>>(Sx, Cw, T);      // T reused as T3
  k_inv_w<<<2048, 256, 163840, stream>>>(T, Cw, bias, out);
}